// CombineRadialSpeciesWithAngular_48060684042913
// MI455X (gfx1250) — compile-verified
//
#include <hip/hip_runtime.h>

typedef __attribute__((ext_vector_type(2))) float v2f;
typedef __attribute__((ext_vector_type(8))) float v8f;

#define N_COMBINED 256
#define KDIM 128
#define LDS_A_PITCH 132   // 128 + 4 pad: conflict-free b64 A-fragment reads

// out[row*256 + col] for the 8 rows covered by one 16x16 f32 accumulator.
// C/D layout: VGPR i -> lanes 0-15: M=i, lanes 16-31: M=8+i; N = lane&15.
__device__ __forceinline__ void store_tile(float* __restrict__ out, const v8f c,
                                           int row_base, int col, int M) {
#pragma unroll
  for (int i = 0; i < 8; ++i) {
    int row = row_base + i;
    if (row < M)
      __builtin_nontemporal_store(c[i], out + (size_t)row * N_COMBINED + col);
  }
}

__global__ __launch_bounds__(256) void combine_gemm_kernel(
    const float* __restrict__ A,   // [M, 128] row-major
    const float* __restrict__ Wl,  // [128, 256] row-major
    float* __restrict__ out,       // [M, 256] row-major (already offset per l)
    int M) {
  __shared__ __attribute__((aligned(16))) float ldsA[64 * LDS_A_PITCH];
  __shared__ __attribute__((aligned(16))) float ldsB[32 * 128 * 4];  // (k4, n, kk)

  const int t    = threadIdx.x;
  const int lane = t & 31;
  const int w    = t >> 5;
  const int l15  = lane & 15;
  const int hi   = lane >> 4;

  const int m_base = blockIdx.x * 64;
  const int n_base = blockIdx.y * 128;

  // ---- Stage A: 64 rows x 128 cols -> LDS (pitch 132). Coalesced b128 loads:
  // each wave reads one full 512B row per iteration.
  #pragma unroll
  for (int i = 0; i < 8; ++i) {
    int row  = i * 8 + w;
    int grow = m_base + row;
    float4 v = make_float4(0.f, 0.f, 0.f, 0.f);
    if (grow < M)
      v = *(const float4*)(A + (size_t)grow * KDIM + lane * 4);
    *(float4*)(&ldsA[row * LDS_A_PITCH + lane * 4]) = v;
  }

  // ---- Stage B slice: W[0:128, n_base:n_base+128] -> LDS swizzled (k4,n,kk)
  // so a B fragment (v0=B[k][n], v1=B[k+1][n] per half-wave K-pair) is one b64.
  // Global reads: 4 coalesced row loads per iteration.
  #pragma unroll
  for (int i = 0; i < 16; ++i) {
    int n  = t & 127;
    int k4 = 2 * i + (t >> 7);
    float4 v;
    v.x = Wl[(k4 * 4 + 0) * N_COMBINED + n_base + n];
    v.y = Wl[(k4 * 4 + 1) * N_COMBINED + n_base + n];
    v.z = Wl[(k4 * 4 + 2) * N_COMBINED + n_base + n];
    v.w = Wl[(k4 * 4 + 3) * N_COMBINED + n_base + n];
    *(float4*)(&ldsB[(k4 * 128 + n) * 4]) = v;
  }

  __syncthreads();

  // ---- Compute: wave (wm, wn) owns rows [rA, rA+32), cols [cB, cB+32)
  const int wm = w & 1;
  const int wn = w >> 1;
  const int rA = wm * 32;
  const int cB = wn * 32;

  v8f acc00 = {}, acc01 = {}, acc10 = {}, acc11 = {};

  // A-fragment: lane -> row (l15), K-pair selected by half-wave (2*hi)
  const float* pa0 = &ldsA[(rA + l15) * LDS_A_PITCH + 2 * hi];
  const float* pa1 = &ldsA[(rA + 16 + l15) * LDS_A_PITCH + 2 * hi];
  // B-fragment: lane -> col (l15), contiguous (kk) pair at 2*hi
  const float* pb0 = &ldsB[(cB + l15) * 4 + 2 * hi];
  const float* pb1 = &ldsB[(cB + 16 + l15) * 4 + 2 * hi];

  #pragma unroll
  for (int k4 = 0; k4 < 32; ++k4) {
    v2f a0 = *(const v2f*)(pa0 + k4 * 4);
    v2f a1 = *(const v2f*)(pa1 + k4 * 4);
    v2f b0 = *(const v2f*)(pb0 + (size_t)k4 * 128 * 4);
    v2f b1 = *(const v2f*)(pb1 + (size_t)k4 * 128 * 4);
    acc00 = __builtin_amdgcn_wmma_f32_16x16x4_f32(false, a0, false, b0,
                                                  (short)0, acc00, false, false);
    acc01 = __builtin_amdgcn_wmma_f32_16x16x4_f32(false, a0, false, b1,
                                                  (short)0, acc01, false, false);
    acc10 = __builtin_amdgcn_wmma_f32_16x16x4_f32(false, a1, false, b0,
                                                  (short)0, acc10, false, false);
    acc11 = __builtin_amdgcn_wmma_f32_16x16x4_f32(false, a1, false, b1,
                                                  (short)0, acc11, false, false);
  }

  // ---- Store (nontemporal: output is streamed once, keep L2 for W/A)
  const int col0 = n_base + cB + l15;
  const int row00 = m_base + rA + hi * 8;
  store_tile(out, acc00, row00,      col0,      M);
  store_tile(out, acc01, row00,      col0 + 16, M);
  store_tile(out, acc10, row00 + 16, col0,      M);
  store_tile(out, acc11, row00 + 16, col0 + 16, M);
}

extern "C" void kernel_launch(void* const* d_in, const int* in_sizes, int n_in,
                              void* d_out, int out_size, void* d_ws, size_t ws_size,
                              hipStream_t stream) {
  (void)in_sizes; (void)n_in; (void)out_size; (void)d_ws; (void)ws_size;
  const float* W = (const float*)d_in[6];   // [6, 128, 256]
  float* out = (float*)d_out;               // [720000, 256]
  size_t row_off = 0;
  for (int l = 0; l < 6; ++l) {
    int M = 20000 * (2 * l + 1);
    dim3 grid((unsigned)((M + 63) / 64), 2, 1);
    combine_gemm_kernel<<<grid, 256, 0, stream>>>(
        (const float*)d_in[l], W + (size_t)l * KDIM * N_COMBINED,
        out + row_off * N_COMBINED, M);
    row_off += (size_t)M;
  }
}